// HalfKAModel_8392366097054
// MI455X (gfx1250) — compile-verified
//
#include <hip/hip_runtime.h>

typedef __attribute__((ext_vector_type(2))) float v2f;
typedef __attribute__((ext_vector_type(8))) float v8f;

#define VSZ 45056
#define DSZ 520
#define BSZ 4096
#define LSZ 32
#define NEXP 8

// ---------------------------------------------------------------------------
// Kernel 1: embedding gather + sum, avg-score head, bucket id.
// One block (256 threads) per batch element. Column-coalesced accumulation.
// ---------------------------------------------------------------------------
__global__ __launch_bounds__(256) void gather_sum_kernel(
    const float* __restrict__ emb_own, const float* __restrict__ emb_opp,
    const float* __restrict__ Wavg, const float* __restrict__ bavg,
    const int* __restrict__ own_idx, const int* __restrict__ opp_idx,
    const int* __restrict__ piece_counts,
    float* __restrict__ xout,      // B x 1024  = [own512 | opp512]
    float* __restrict__ avg_out,   // B
    int* __restrict__ bucket_out)  // B
{
    const int b   = blockIdx.x;
    const int tid = threadIdx.x;

    __shared__ int   rows[2 * LSZ];
    __shared__ float diff8[8];

    if (tid < LSZ)            rows[tid] = own_idx[b * LSZ + tid];
    else if (tid < 2 * LSZ)   rows[tid] = opp_idx[b * LSZ + (tid - LSZ)];
    __syncthreads();

    for (int c = tid; c < DSZ; c += 256) {
        float so = 0.0f, sp = 0.0f;
#pragma unroll 8
        for (int l = 0; l < LSZ; ++l) {
            so += emb_own[(size_t)rows[l]       * DSZ + c];
            sp += emb_opp[(size_t)rows[LSZ + l] * DSZ + c];
        }
        if (c < 8) {
            diff8[c] = so - sp;
        } else {
            xout[(size_t)b * 1024 + (c - 8)]       = so;
            xout[(size_t)b * 1024 + 512 + (c - 8)] = sp;
        }
    }
    __syncthreads();

    if (tid == 0) {
        float a = bavg[0];
#pragma unroll
        for (int i = 0; i < 8; ++i) a += diff8[i] * Wavg[i];
        avg_out[b] = a;

        int pc = piece_counts[b];
        int bk = (pc - 1) / 4;           // pc==0 -> 0 after clamp (matches floor+clip)
        bk = bk < 0 ? 0 : (bk > NEXP - 1 ? NEXP - 1 : bk);
        bucket_out[b] = bk;
    }
}

// ---------------------------------------------------------------------------
// Kernel 2: deterministic per-bucket compaction (one wave32 per bucket).
// ---------------------------------------------------------------------------
__global__ __launch_bounds__(32) void bucket_build_kernel(
    const int* __restrict__ bucket, int* __restrict__ lists, int* __restrict__ counts)
{
    const int e    = blockIdx.x;
    const int lane = threadIdx.x;
    int base = 0;
    for (int i0 = 0; i0 < BSZ; i0 += 32) {
        const int  b = i0 + lane;
        const bool m = (bucket[b] == e);
        const unsigned mask = (unsigned)__ballot(m);
        const int pre = __popc(mask & ((1u << lane) - 1u));
        if (m) lists[e * BSZ + base + pre] = b;
        base += __popc(mask);
    }
    if (lane == 0) counts[e] = base;
}

// ---------------------------------------------------------------------------
// Kernel 3: per-bucket MLP. One wave per 16-row tile.
// Layer 1 (1024 -> 16) via V_WMMA_F32_16X16X4_F32 with W1[e] staged in LDS.
// Layers 2/3 on lanes 0..15 (trivial FLOPs).
// ---------------------------------------------------------------------------
__global__ __launch_bounds__(32) void mlp_wmma_kernel(
    const float* __restrict__ x, const float* __restrict__ avg,
    const int* __restrict__ lists, const int* __restrict__ counts,
    const float* __restrict__ W1, const float* __restrict__ b1,
    const float* __restrict__ W2, const float* __restrict__ b2,
    const float* __restrict__ W3, const float* __restrict__ b3,
    float* __restrict__ out)
{
    const int e = blockIdx.y;
    const int t = blockIdx.x;
    const int n = counts[e];
    if (t * 16 >= n) return;
    const int lane = threadIdx.x;

    __shared__ float w1s[16 * 1024];   // 64 KB: W1[e], row o = output index

    {
        const float4* src = (const float4*)(W1 + (size_t)e * 16 * 1024);
        float4*       dst = (float4*)w1s;
        for (int i = lane; i < 4096; i += 32) dst[i] = src[i];
    }
    __syncthreads();

    // A fragment (16x4 f32): lanes 0-15 hold M=lane, K={k0,k0+1}; lanes 16-31 K={k0+2,k0+3}
    const int m  = lane & 15;            // A row / B column (N) / layer-1 output idx
    const int kb = (lane >> 4) * 2;      // K sub-pair select
    const int r  = t * 16 + m;
    const int rc = r < n ? r : (n - 1);  // pad with last valid row (discarded at store)
    const int g  = lists[e * BSZ + rc];

    const float2* xr2 = (const float2*)(x + (size_t)g * 1024 + kb);
    const float2* wr2 = (const float2*)(w1s + m * 1024 + kb);

    v8f c = {0.f, 0.f, 0.f, 0.f, 0.f, 0.f, 0.f, 0.f};
#pragma unroll 4
    for (int k0 = 0; k0 < 1024; k0 += 4) {
        const float2 av = xr2[k0 >> 1];
        const float2 bv = wr2[k0 >> 1];
        v2f af; af.x = av.x; af.y = av.y;
        v2f bf; bf.x = bv.x; bf.y = bv.y;
        c = __builtin_amdgcn_wmma_f32_16x16x4_f32(
                /*neg_a=*/false, af, /*neg_b=*/false, bf,
                /*c_mod=*/(short)0, c, /*reuse_a=*/false, /*reuse_b=*/false);
    }

    __syncthreads();
    float* h1s = w1s;                    // reuse LDS: 16x16 h1 tile
    {
        const float bb = b1[e * 16 + m];
        const int   mh = (lane >> 4) * 8;   // C layout: lanes>=16 hold rows M+8
#pragma unroll
        for (int j = 0; j < 8; ++j) {
            float v = c[j] + bb;
            v = v < 0.f ? 0.f : (v > 1.f ? 1.f : v);
            h1s[(mh + j) * 16 + m] = v;
        }
    }
    __syncthreads();

    if (lane < 16 && (t * 16 + lane) < n) {
        const int gg = lists[e * BSZ + t * 16 + lane];
        const float* h1r = h1s + lane * 16;
        const float* w2  = W2 + (size_t)e * 32 * 16;
        const float* w3  = W3 + (size_t)e * 32;
        float acc = 0.f;
#pragma unroll 4
        for (int o = 0; o < 32; ++o) {
            float s = b2[e * 32 + o];
#pragma unroll
            for (int i = 0; i < 16; ++i) s += w2[o * 16 + i] * h1r[i];
            s = s < 0.f ? 0.f : (s > 1.f ? 1.f : s);
            acc += w3[o] * s;
        }
        out[gg] = acc + b3[e] + avg[gg];
    }
}

// ---------------------------------------------------------------------------
extern "C" void kernel_launch(void* const* d_in, const int* in_sizes, int n_in,
                              void* d_out, int out_size, void* d_ws, size_t ws_size,
                              hipStream_t stream)
{
    const float* emb_own = (const float*)d_in[0];
    const float* emb_opp = (const float*)d_in[1];
    const float* Wavg    = (const float*)d_in[2];
    const float* bavg    = (const float*)d_in[3];
    const float* W1      = (const float*)d_in[4];
    const float* b1      = (const float*)d_in[5];
    const float* W2      = (const float*)d_in[6];
    const float* b2      = (const float*)d_in[7];
    const float* W3      = (const float*)d_in[8];
    const float* b3      = (const float*)d_in[9];
    const int* own_idx   = (const int*)d_in[10];
    const int* opp_idx   = (const int*)d_in[11];
    const int* pieces    = (const int*)d_in[12];
    float* out = (float*)d_out;

    char* ws = (char*)d_ws;
    size_t off = 0;
    float* xbuf   = (float*)(ws + off); off += (size_t)BSZ * 1024 * sizeof(float); // 16 MB
    float* avgb   = (float*)(ws + off); off += (size_t)BSZ * sizeof(float);
    int*   bucket = (int*)  (ws + off); off += (size_t)BSZ * sizeof(int);
    int*   lists  = (int*)  (ws + off); off += (size_t)NEXP * BSZ * sizeof(int);
    int*   counts = (int*)  (ws + off); off += (size_t)NEXP * sizeof(int);

    hipLaunchKernelGGL(gather_sum_kernel, dim3(BSZ), dim3(256), 0, stream,
                       emb_own, emb_opp, Wavg, bavg, own_idx, opp_idx, pieces,
                       xbuf, avgb, bucket);
    hipLaunchKernelGGL(bucket_build_kernel, dim3(NEXP), dim3(32), 0, stream,
                       bucket, lists, counts);
    hipLaunchKernelGGL(mlp_wmma_kernel, dim3(BSZ / 16, NEXP), dim3(32), 0, stream,
                       xbuf, avgb, lists, counts, W1, b1, W2, b2, W3, b3, out);
}